// Huber_89266600280318
// MI455X (gfx1250) — compile-verified
//
#include <hip/hip_runtime.h>
#include <math.h>

typedef __attribute__((ext_vector_type(2))) float v2f;
typedef __attribute__((ext_vector_type(4))) float v4f;
typedef __attribute__((ext_vector_type(8))) float v8f;

#define N4K 4096
#define RR 10     // logical rank
#define RS 16     // padded row stride for D/beta (zeros in cols 10..15)
#define PDIM 112  // logical Gram width: 100 outer-product cols + nobs col + pad
#define PV 128    // P stored lane-major: [p][16 lanes][8 tiles]

static __device__ __forceinline__ v8f wmma_f32x4(v2f a, v2f b, v8f c) {
    // V_WMMA_F32_16X16X4_F32 : D = A(16x4) * B(4x16) + C(16x16)
    return __builtin_amdgcn_wmma_f32_16x16x4_f32(false, a, false, b, (short)0, c, false, false);
}

// ---------------------------------------------------------------- alpha ----
__global__ void k_alpha(const float* c_p, float* alpha_ws) {
    float c = *c_p;
    // chi2_cdf(c^2,1) = erf(c/sqrt(2))
    // chi2_cdf(c^2,3) = erf(c/sqrt(2)) - sqrt(2/pi)*c*exp(-c^2/2)
    float e = erff(c * 0.70710678118654752f);
    float cdf3 = e - 0.79788456080286536f * c * expf(-0.5f * c * c);
    *alpha_ws = 0.5f * c * c * (1.0f - e) + 0.5f * cdf3;
}

// ------------------------------------- init Ub, Vb (stride-16, zero pad) ---
__global__ void k_init(const float* U, const float* V, float* Ub, float* Vb) {
    int idx = blockIdx.x * blockDim.x + threadIdx.x;
    if (idx < N4K * RS) {
        int j = idx >> 4, k = idx & 15;
        Ub[idx] = (k < RR) ? U[j * RR + k] : 0.f;            // U is [4096,10]
        Vb[idx] = (k < RR) ? V[(long)k * N4K + j] : 0.f;     // Vb = V.T
    }
}

// ------------------- P lane-major: P[p][n][t] = outer(D[p])[t*16+n] -------
__global__ void k_pmat(const float* D, float* P) {
    int idx = blockIdx.x * blockDim.x + threadIdx.x;
    if (idx >= N4K * PV) return;
    int p = idx >> 7, rem = idx & 127;
    int n = rem >> 3, t = rem & 7;
    int col = t * 16 + n;                      // logical Gram column
    float v = 0.f;
    if (t < 7) {
        if (col < 100)       v = D[p * RS + col / RR] * D[p * RS + col % RR];
        else if (col == 100) v = 1.0f;         // -> nobs column
    }
    P[idx] = v;
}

// --------------------------------- Gram: Gpart[ch] += mask.T @ P ----------
__global__ void __launch_bounds__(128)
k_gram(const float* X, const float* P, float* Gpart, long rs, long cs) {
    int tid = threadIdx.x;
    int wave = tid >> 5, lane = tid & 31;
    int half = lane >> 4, l16 = lane & 15;
    int q0 = (blockIdx.x * 4 + wave) * 16;
    int pbase = blockIdx.y * 1024;             // K-chunk of 1024
    int j = q0 + l16;

    v8f z = {0.f,0.f,0.f,0.f,0.f,0.f,0.f,0.f};
    v8f acc[7];
#pragma unroll
    for (int t = 0; t < 7; ++t) acc[t] = z;

    for (int p0 = pbase; p0 < pbase + 1024; p0 += 4) {
        int pa = p0 + 2 * half;                // this half's K rows
        float x0 = X[(long)pa * rs + (long)j * cs];
        float x1 = X[(long)(pa + 1) * rs + (long)j * cs];
        __builtin_prefetch(&X[(long)(pa + 4) * rs + (long)j * cs], 0, 1);
        v2f a = { (x0 != 0.0f) ? 1.0f : 0.0f,
                  (x1 != 0.0f) ? 1.0f : 0.0f };
        const float* Pb = P + (long)pa * PV + l16 * 8;   // 8 contiguous per lane
        v4f r0a = *(const v4f*)(Pb);
        v4f r0b = *(const v4f*)(Pb + 4);
        v4f r1a = *(const v4f*)(Pb + PV);
        v4f r1b = *(const v4f*)(Pb + PV + 4);
#pragma unroll
        for (int t = 0; t < 7; ++t) {
            v2f b = { (t < 4) ? r0a[t] : r0b[t - 4],
                      (t < 4) ? r1a[t] : r1b[t - 4] };
            acc[t] = wmma_f32x4(a, b, acc[t]);
        }
    }
    long base = ((long)blockIdx.y * N4K + q0) * PDIM;
#pragma unroll
    for (int v = 0; v < 8; ++v) {
        int row = v + 8 * half;
#pragma unroll
        for (int t = 0; t < 7; ++t)
            Gpart[base + (long)row * PDIM + t * 16 + l16] = acc[t][v];
    }
}

// ------------------------- reduce Gram, invert 10x10, reset sigma ---------
__global__ void k_inv(const float* Gpart, float* Ginv, float* nobsb,
                      float* sigma, const float* sigma0_p) {
    int j = blockIdx.x * blockDim.x + threadIdx.x;
    if (j >= N4K) return;
    float A[100], B[100];
    const float* g = Gpart + (long)j * PDIM;
    for (int t = 0; t < 100; ++t) {
        float s = 0.f;
        for (int ch = 0; ch < 4; ++ch) s += g[(long)ch * N4K * PDIM + t];
        A[t] = s;
    }
    float nobs = 0.f;
    for (int ch = 0; ch < 4; ++ch) nobs += g[(long)ch * N4K * PDIM + 100];
    for (int t = 0; t < 100; ++t) B[t] = 0.f;
    for (int k = 0; k < RR; ++k) B[k * RR + k] = 1.f;
    // Gauss-Jordan (G is an SPD Gram matrix -> no pivoting)
    for (int k = 0; k < RR; ++k) {
        float piv = 1.0f / A[k * RR + k];
        for (int col = 0; col < RR; ++col) { A[k*RR+col] *= piv; B[k*RR+col] *= piv; }
        for (int i = 0; i < RR; ++i) {
            if (i == k) continue;
            float f = A[i * RR + k];
            for (int col = 0; col < RR; ++col) {
                A[i*RR+col] -= f * A[k*RR+col];
                B[i*RR+col] -= f * B[k*RR+col];
            }
        }
    }
    for (int t = 0; t < 100; ++t) Ginv[(long)j * 100 + t] = B[t];
    nobsb[j] = nobs;
    sigma[j] = *sigma0_p;
}

// ------------- pass1: ssum_part[ch][j] = sum_p psi((Y-Db)/sigma)^2 --------
__global__ void __launch_bounds__(128)
k_pass1(const float* X, const float* D, const float* beta,
        const float* sigma, const float* c_p, float* ssp, long rs, long cs) {
    int tid = threadIdx.x;
    int wave = tid >> 5, lane = tid & 31;
    int half = lane >> 4, l16 = lane & 15;
    int q0 = (blockIdx.x * 4 + wave) * 16;
    int pbase = blockIdx.y * 512;
    int j = q0 + l16;
    float sig = sigma[j];
    float cc = *c_p;

    // B = beta^T (12x16): per-lane column j fixed, rows zero-padded past 10
    const float* brp = beta + (long)j * RS;
    v2f b0 = *(const v2f*)(brp + 2 * half);
    v2f b1 = *(const v2f*)(brp + 4 + 2 * half);
    v2f b2 = *(const v2f*)(brp + 8 + 2 * half);    // half==1 -> zeros (pad)

    float acc = 0.f;
    v8f z = {0.f,0.f,0.f,0.f,0.f,0.f,0.f,0.f};
    for (int p = pbase; p < pbase + 512; p += 16) {
        const float* Dr = D + (long)(p + l16) * RS;
        v2f a0 = *(const v2f*)(Dr + 2 * half);
        v2f a1 = *(const v2f*)(Dr + 4 + 2 * half);
        v2f a2 = *(const v2f*)(Dr + 8 + 2 * half); // half==1 -> zeros (pad)
        v8f r = z;
        r = wmma_f32x4(a0, b0, r);
        r = wmma_f32x4(a1, b1, r);
        r = wmma_f32x4(a2, b2, r);                 // Rhat tile = D @ beta^T
        __builtin_prefetch(&X[(long)(p + 16 + 8 * half) * rs + (long)j * cs], 0, 1);
#pragma unroll
        for (int v = 0; v < 8; ++v) {
            int prow = p + v + 8 * half;
            float x = X[(long)prow * rs + (long)j * cs];
            float rr = (x != 0.0f) ? (x - r[v]) : 0.0f;
            float t = fminf(fmaxf(rr / sig, -cc), cc);
            acc += t * t;
        }
    }
    acc += __shfl_down(acc, 16);
    if (half == 0) ssp[(long)blockIdx.y * N4K + j] = acc;
}

// ---------------------------- sigma *= tau^lamda --------------------------
__global__ void k_sigup(const float* ssp, const float* nobsb, const float* alpha_ws,
                        const float* lam_p, float* sigma) {
    int j = blockIdx.x * blockDim.x + threadIdx.x;
    if (j >= N4K) return;
    float s = 0.f;
    for (int ch = 0; ch < 8; ++ch) s += ssp[(long)ch * N4K + j];
    float tau = sqrtf(s) / sqrtf(2.0f * nobsb[j] * (*alpha_ws));
    sigma[j] *= powf(tau, *lam_p);
}

// -------- pass2: rhs_part[ch] = D^T @ (psi(R/sig)*sig) over p-chunk -------
__global__ void __launch_bounds__(128)
k_pass2(const float* X, const float* D, const float* beta,
        const float* sigma, const float* c_p, float* rhsp, long rs, long cs) {
    __shared__ float Dl[512 * RS];     // 32 KB (<< 320 KB/WGP)
    int tid = threadIdx.x;
    int wave = tid >> 5, lane = tid & 31;
    int half = lane >> 4, l16 = lane & 15;
    int pbase = blockIdx.y * 512;
    for (int idx = tid; idx < 512 * RS; idx += 128)
        Dl[idx] = D[(long)pbase * RS + idx];       // includes zero pad cols
    __syncthreads();

    int q0 = (blockIdx.x * 4 + wave) * 16;
    int j = q0 + l16;
    float sig = sigma[j];
    float cc = *c_p;
    const float* brp = beta + (long)j * RS;
    v4f vb0 = *(const v4f*)(brp);
    v4f vb1 = *(const v4f*)(brp + 4);
    v4f vb2 = *(const v4f*)(brp + 8);              // cols 10..15 are zero

    v8f racc = {0.f,0.f,0.f,0.f,0.f,0.f,0.f,0.f};
    for (int prel = 0; prel < 512; prel += 4) {
        int pr = prel + 2 * half;                  // this half's two K rows
        float zv[2];
#pragma unroll
        for (int s = 0; s < 2; ++s) {
            const float* Drow = &Dl[(pr + s) * RS];
            v4f q0v = *(const v4f*)(Drow);
            v4f q1v = *(const v4f*)(Drow + 4);
            v4f q2v = *(const v4f*)(Drow + 8);
            float rhat = q0v.x*vb0.x + q0v.y*vb0.y + q0v.z*vb0.z + q0v.w*vb0.w
                       + q1v.x*vb1.x + q1v.y*vb1.y + q1v.z*vb1.z + q1v.w*vb1.w
                       + q2v.x*vb2.x + q2v.y*vb2.y;       // pads are zero
            float x = X[(long)(pbase + pr + s) * rs + (long)j * cs];
            float rr = (x != 0.0f) ? (x - rhat) : 0.0f;
            float t = fminf(fmaxf(rr / sig, -cc), cc);
            zv[s] = t * sig;
        }
        v2f b = { zv[0], zv[1] };                  // Z 4x16 fragment
        v2f a = { Dl[pr * RS + l16],               // D^T 16x4 (rows>=10 zero)
                  Dl[(pr + 1) * RS + l16] };
        racc = wmma_f32x4(a, b, racc);
    }
#pragma unroll
    for (int v = 0; v < 8; ++v) {
        int k = v + 8 * half;
        if (k < RR) rhsp[((long)blockIdx.y * 16 + k) * N4K + j] = racc[v];
    }
}

// ------------------- beta += mu * (Ginv @ rhs) per regression -------------
__global__ void k_betaup(const float* rhsp, const float* Ginv, const float* mu_p,
                         float* beta) {
    int j = blockIdx.x * blockDim.x + threadIdx.x;
    if (j >= N4K) return;
    float r[RR];
    for (int k = 0; k < RR; ++k) {
        float s = 0.f;
        for (int ch = 0; ch < 8; ++ch) s += rhsp[((long)ch * 16 + k) * N4K + j];
        r[k] = s;
    }
    const float* gi = Ginv + (long)j * 100;
    float mu = *mu_p;
    for (int i = 0; i < RR; ++i) {
        float d = 0.f;
        for (int k = 0; k < RR; ++k) d += gi[i * RR + k] * r[k];
        beta[(long)j * RS + i] += mu * d;
    }
}

// ----------------------------- out = Ub @ Vb^T ----------------------------
__global__ void __launch_bounds__(128)
k_out(const float* Ub, const float* Vb, float* out) {
    int tid = threadIdx.x;
    int wave = tid >> 5, lane = tid & 31;
    int half = lane >> 4, l16 = lane & 15;
    int i0 = (blockIdx.x * 4 + wave) * 16;
    int j0 = blockIdx.y * 16;

    const float* vr = Vb + (long)(j0 + l16) * RS;
    v2f b0 = *(const v2f*)(vr + 2 * half);
    v2f b1 = *(const v2f*)(vr + 4 + 2 * half);
    v2f b2 = *(const v2f*)(vr + 8 + 2 * half);
    const float* ur = Ub + (long)(i0 + l16) * RS;
    v2f a0 = *(const v2f*)(ur + 2 * half);
    v2f a1 = *(const v2f*)(ur + 4 + 2 * half);
    v2f a2 = *(const v2f*)(ur + 8 + 2 * half);

    v8f r = {0.f,0.f,0.f,0.f,0.f,0.f,0.f,0.f};
    r = wmma_f32x4(a0, b0, r);
    r = wmma_f32x4(a1, b1, r);
    r = wmma_f32x4(a2, b2, r);
#pragma unroll
    for (int v = 0; v < 8; ++v)
        out[(long)(i0 + v + 8 * half) * N4K + j0 + l16] = r[v];
}

// ==========================================================================
extern "C" void kernel_launch(void* const* d_in, const int* in_sizes, int n_in,
                              void* d_out, int out_size, void* d_ws, size_t ws_size,
                              hipStream_t stream) {
    const float* U     = (const float*)d_in[0];
    const float* V     = (const float*)d_in[1];
    const float* X     = (const float*)d_in[2];
    const float* c_p   = (const float*)d_in[3];
    const float* lam_p = (const float*)d_in[4];
    const float* mu_p  = (const float*)d_in[5];
    const float* sig0  = (const float*)d_in[6];
    float* out = (float*)d_out;

    float* w        = (float*)d_ws;
    float* alpha_ws = w;                               // 1 (+pad to 8)
    float* sigma    = w + 8;                           // 4096
    float* Ub       = sigma + N4K;                     // 4096*16
    float* Vb       = Ub + (long)N4K * RS;             // 4096*16
    float* Pbuf     = Vb + (long)N4K * RS;             // 4096*128
    float* Gpart    = Pbuf + (long)N4K * PV;           // 4*4096*112
    float* Ginv     = Gpart + 4L * N4K * PDIM;         // 4096*100
    float* nobsb    = Ginv + (long)N4K * 100;          // 4096
    float* ssp      = nobsb + N4K;                     // 8*4096
    float* rhsp     = ssp + 8L * N4K;                  // 8*16*4096
    (void)in_sizes; (void)n_in; (void)out_size; (void)ws_size;

    k_alpha<<<dim3(1), dim3(1), 0, stream>>>(c_p, alpha_ws);
    k_init<<<dim3((N4K * RS + 255) / 256), dim3(256), 0, stream>>>(U, V, Ub, Vb);

    auto hubreg = [&](float* beta, const float* D, long rs, long cs) {
        k_pmat<<<dim3((N4K * PV + 255) / 256), dim3(256), 0, stream>>>(D, Pbuf);
        k_gram<<<dim3(64, 4), dim3(128), 0, stream>>>(X, Pbuf, Gpart, rs, cs);
        k_inv<<<dim3(16), dim3(256), 0, stream>>>(Gpart, Ginv, nobsb, sigma, sig0);
        for (int it = 0; it < 2; ++it) {
            k_pass1<<<dim3(64, 8), dim3(128), 0, stream>>>(X, D, beta, sigma, c_p, ssp, rs, cs);
            k_sigup<<<dim3(16), dim3(256), 0, stream>>>(ssp, nobsb, alpha_ws, lam_p, sigma);
            k_pass2<<<dim3(64, 8), dim3(128), 0, stream>>>(X, D, beta, sigma, c_p, rhsp, rs, cs);
            k_betaup<<<dim3(16), dim3(256), 0, stream>>>(rhsp, Ginv, mu_p, beta);
        }
    };

    for (int layer = 0; layer < 3; ++layer) {
        hubreg(Vb, Ub, (long)N4K, 1L);   // V-step: Y(p,q) = X[p,q]
        hubreg(Ub, Vb, 1L, (long)N4K);   // U-step: Y(p,q) = X[q,p]
    }

    k_out<<<dim3(64, 256), dim3(128), 0, stream>>>(Ub, Vb, out);
}